// S4Checkpointed_12421045420509
// MI455X (gfx1250) — compile-verified
//
#include <hip/hip_runtime.h>
#include <math.h>

typedef __attribute__((ext_vector_type(2))) float v2f;
typedef __attribute__((ext_vector_type(8))) float v8f;
typedef __attribute__((ext_vector_type(4))) int   v4i;

#define D_INN   768
#define NSTATE  16
#define DTRANK  48
#define BATCH   2
#define SEQLEN  2048
#define NPROJ   80            // 16 (B) + 16 (C) + 48 (dt)
#define LC      32            // timesteps per chunk
#define NCHUNK  64            // SEQLEN / LC
#define M_TOTAL (BATCH*SEQLEN) // 4096

// ---- CDNA5 async global->LDS (guarded; falls back to synchronous staging) ----
#if defined(__has_builtin)
#if __has_builtin(__builtin_amdgcn_global_load_async_to_lds_b128)
#define HAVE_ASYNC_LDS 1
#endif
#endif
#ifndef HAVE_ASYNC_LDS
#define HAVE_ASYNC_LDS 0
#endif

#if HAVE_ASYNC_LDS
typedef __attribute__((address_space(1))) v4i* g_v4i_p;   // global (device) AS
typedef __attribute__((address_space(3))) v4i* l_v4i_p;   // LDS AS
__device__ __forceinline__ void async_lds_b128(const float* gsrc, float* ldst) {
    __builtin_amdgcn_global_load_async_to_lds_b128(
        (g_v4i_p)(void*)gsrc, (l_v4i_p)(void*)ldst, 0, 0);
}
__device__ __forceinline__ void wait_async0() {
#if __has_builtin(__builtin_amdgcn_s_wait_asynccnt)
    __builtin_amdgcn_s_wait_asynccnt(0);
#else
    asm volatile("s_wait_asynccnt 0x0" ::: "memory");
#endif
}
#endif

// ---------------------------------------------------------------------------
// Kernel 1: P80 = X(4096x768) @ Wsub^T(768x80) via V_WMMA_F32_16X16X4_F32.
// Only the 80 live rows of W_xproj are touched:
//   out col j -> W row: j<16 ? j : (j<32 ? j+16 : j+1552)
// One wave per 16-row M tile; 5 accumulators cover all 80 output columns,
// so X is read exactly once from HBM (W stays L2-resident, 245 KB).
// ---------------------------------------------------------------------------
__global__ __launch_bounds__(32)
void k_proj_gemm(const float* __restrict__ X,
                 const float* __restrict__ W,
                 float* __restrict__ P80)
{
    const int m0   = blockIdx.x * 16;
    const int lane = threadIdx.x;       // 0..31 (wave32)
    const int half = lane >> 4;         // K-half select
    const int idx  = lane & 15;         // M (for A) / N (for B,D)

    // A-frag: 16x4 f32, lane<16 -> M=lane holds K=0,1; lane>=16 -> K=2,3
    const float* arow = X + (size_t)(m0 + idx) * D_INN + 2 * half;

    const float* brow[5];
#pragma unroll
    for (int t = 0; t < 5; ++t) {
        int j    = t * 16 + idx;
        int wrow = (j < 16) ? j : ((j < 32) ? (j + 16) : (j + 1552));
        brow[t]  = W + (size_t)wrow * D_INN + 2 * half;
    }

    v8f acc[5];
    const v8f vzero = {0.f,0.f,0.f,0.f,0.f,0.f,0.f,0.f};
#pragma unroll
    for (int t = 0; t < 5; ++t) acc[t] = vzero;

    for (int k = 0; k < D_INN; k += 4) {
        v2f a = *(const v2f*)(arow + k);
#pragma unroll
        for (int t = 0; t < 5; ++t) {
            v2f b = *(const v2f*)(brow[t] + k);
            acc[t] = __builtin_amdgcn_wmma_f32_16x16x4_f32(
                false, a, false, b, (short)0, acc[t], false, false);
        }
    }

    // D layout: VGPR r, lane l -> row m0 + r + 8*(l>>4), col (l&15)
#pragma unroll
    for (int t = 0; t < 5; ++t) {
        float* o = P80 + t * 16 + idx;
#pragma unroll
        for (int r = 0; r < 8; ++r)
            o[(size_t)(m0 + r + 8 * half) * NPROJ] = acc[t][r];
    }
}

// ---------------------------------------------------------------------------
// Kernel 2: Delta = softplus(dtraw(4096x48) @ W_dt^T(48x768) + 2*b_dt), f32 WMMA.
// dtraw lives in P80 columns 32..79. Epilogue fuses bias + softplus.
// ---------------------------------------------------------------------------
__global__ __launch_bounds__(32)
void k_delta_gemm(const float* __restrict__ P80,
                  const float* __restrict__ Wdt,
                  const float* __restrict__ bdt,
                  float* __restrict__ Delta)
{
    const int m0   = blockIdx.x * 16;
    const int c0   = blockIdx.y * 16;
    const int lane = threadIdx.x;
    const int half = lane >> 4;
    const int idx  = lane & 15;

    const float* arow = P80 + (size_t)(m0 + idx) * NPROJ + 32 + 2 * half;
    const float* brow = Wdt + (size_t)(c0 + idx) * DTRANK + 2 * half;

    v8f acc = {0.f,0.f,0.f,0.f,0.f,0.f,0.f,0.f};
#pragma unroll
    for (int k = 0; k < DTRANK; k += 4) {
        v2f a = *(const v2f*)(arow + k);
        v2f b = *(const v2f*)(brow + k);
        acc = __builtin_amdgcn_wmma_f32_16x16x4_f32(
            false, a, false, b, (short)0, acc, false, false);
    }

    const float bias = 2.0f * bdt[c0 + idx];
    float* o = Delta + c0 + idx;
#pragma unroll
    for (int r = 0; r < 8; ++r) {
        float v  = acc[r] + bias;
        float sp = (v > 20.0f) ? v : log1pf(__expf(v));   // softplus
        o[(size_t)(m0 + r + 8 * half) * D_INN] = sp;
    }
}

// ---------------------------------------------------------------------------
// Phase A: per-chunk local scan with h=0, accumulating the chunk decay
// product P[n] = prod_t exp(delta_t * A[n]) and local end state h[n].
// One thread per channel (16 states in VGPRs); Bm staged in LDS via
// GLOBAL_LOAD_ASYNC_TO_LDS_B128 when available.
// ---------------------------------------------------------------------------
__global__ __launch_bounds__(D_INN)
void k_scan_local(const float* __restrict__ X,
                  const float* __restrict__ Delta,
                  const float* __restrict__ P80,
                  const float* __restrict__ A_log,
                  float* __restrict__ Pprod,
                  float* __restrict__ Hend)
{
    __shared__ __align__(16) float sB[LC * NSTATE];   // 2 KB
    const int ch = threadIdx.x;
    const int c  = blockIdx.x;
    const int b  = blockIdx.y;
    const size_t row0 = (size_t)b * SEQLEN + (size_t)c * LC;

#if HAVE_ASYNC_LDS
    if (ch < (LC * NSTATE) / 4) {                     // 128 lanes, 16B each
        const int i = ch * 4;                         // stays inside one 16-col row
        async_lds_b128(&P80[(row0 + (i >> 4)) * NPROJ + (i & 15)], &sB[i]);
    }
#endif

    float Aa[NSTATE];
#pragma unroll
    for (int n = 0; n < NSTATE; ++n)
        Aa[n] = -__expf(A_log[ch * NSTATE + n]);

#if HAVE_ASYNC_LDS
    wait_async0();
#else
    for (int i = ch; i < LC * NSTATE; i += blockDim.x)
        sB[i] = P80[(row0 + (i >> 4)) * NPROJ + (i & 15)];
#endif
    __syncthreads();

    float h[NSTATE], P[NSTATE];
#pragma unroll
    for (int n = 0; n < NSTATE; ++n) { h[n] = 0.f; P[n] = 1.f; }

    for (int t = 0; t < LC; ++t) {
        const size_t off = (row0 + t) * D_INN + ch;
        if (t + 8 < LC) {                              // stream-ahead hint
            __builtin_prefetch(&Delta[off + 8 * D_INN], 0, 0);
            __builtin_prefetch(&X[off + 8 * D_INN], 0, 0);
        }
        const float delta = Delta[off];
        const float u     = delta * X[off];
        const float4* Bv  = (const float4*)(sB + t * NSTATE);
#pragma unroll
        for (int q = 0; q < 4; ++q) {
            float4 bq = Bv[q];
            float bb[4] = {bq.x, bq.y, bq.z, bq.w};
#pragma unroll
            for (int j = 0; j < 4; ++j) {
                const int n = q * 4 + j;
                const float dA = __expf(delta * Aa[n]);
                h[n] = dA * h[n] + u * bb[j];
                P[n] *= dA;
            }
        }
    }

    const size_t so = (((size_t)b * NCHUNK + c) * D_INN + ch) * NSTATE;
    float4* Pp = (float4*)(Pprod + so);
    float4* Hp = (float4*)(Hend  + so);
#pragma unroll
    for (int q = 0; q < 4; ++q) {
        Pp[q] = make_float4(P[q*4], P[q*4+1], P[q*4+2], P[q*4+3]);
        Hp[q] = make_float4(h[q*4], h[q*4+1], h[q*4+2], h[q*4+3]);
    }
}

// ---------------------------------------------------------------------------
// Phase B: sequential combine over chunk boundaries (64 steps, cheap).
// Hin[c] = incoming state of chunk c;  h_{c} = P_c * h_{c-1} + Hend_c.
// One thread per (b, ch, n) state element.
// ---------------------------------------------------------------------------
__global__ __launch_bounds__(256)
void k_scan_combine(const float* __restrict__ Pprod,
                    const float* __restrict__ Hend,
                    float* __restrict__ Hin)
{
    const int gid = blockIdx.x * blockDim.x + threadIdx.x;   // < BATCH*D_INN*NSTATE
    const int b   = gid / (D_INN * NSTATE);
    const int rem = gid - b * (D_INN * NSTATE);
    float h = 0.f;
    for (int c = 0; c < NCHUNK; ++c) {
        const size_t off = ((size_t)(b * NCHUNK + c)) * (D_INN * NSTATE) + rem;
        Hin[off] = h;
        h = Pprod[off] * h + Hend[off];
    }
}

// ---------------------------------------------------------------------------
// Phase C: re-run each chunk with the correct incoming state, compute
// y_t = sum_n h_t[n]*C_t[n] + x*D, then gate with silu(z^T) and store
// (non-temporal -- output is streamed once).
// ---------------------------------------------------------------------------
__global__ __launch_bounds__(D_INN)
void k_scan_final(const float* __restrict__ X,
                  const float* __restrict__ Delta,
                  const float* __restrict__ P80,
                  const float* __restrict__ A_log,
                  const float* __restrict__ Dvec,
                  const float* __restrict__ Z,      // (b, d, l)
                  const float* __restrict__ Hin,
                  float* __restrict__ Out)          // (b, l, d)
{
    __shared__ __align__(16) float sB[LC * NSTATE];
    __shared__ __align__(16) float sC[LC * NSTATE];
    const int ch = threadIdx.x;
    const int c  = blockIdx.x;
    const int b  = blockIdx.y;
    const size_t row0 = (size_t)b * SEQLEN + (size_t)c * LC;

#if HAVE_ASYNC_LDS
    if (ch < (LC * NSTATE) / 4) {                     // lanes 0..127 -> sB
        const int i = ch * 4;
        async_lds_b128(&P80[(row0 + (i >> 4)) * NPROJ + (i & 15)], &sB[i]);
    } else if (ch < (LC * NSTATE) / 2) {              // lanes 128..255 -> sC
        const int i = (ch - (LC * NSTATE) / 4) * 4;
        async_lds_b128(&P80[(row0 + (i >> 4)) * NPROJ + 16 + (i & 15)], &sC[i]);
    }
#endif

    float Aa[NSTATE];
#pragma unroll
    for (int n = 0; n < NSTATE; ++n)
        Aa[n] = -__expf(A_log[ch * NSTATE + n]);

#if HAVE_ASYNC_LDS
    wait_async0();
#else
    for (int i = ch; i < LC * NSTATE; i += blockDim.x) {
        const size_t pr = (row0 + (i >> 4)) * NPROJ;
        sB[i] = P80[pr + (i & 15)];
        sC[i] = P80[pr + 16 + (i & 15)];
    }
#endif
    __syncthreads();

    float h[NSTATE];
    const size_t so = (((size_t)b * NCHUNK + c) * D_INN + ch) * NSTATE;
    const float4* Hp = (const float4*)(Hin + so);
#pragma unroll
    for (int q = 0; q < 4; ++q) {
        float4 hq = Hp[q];
        h[q*4+0] = hq.x; h[q*4+1] = hq.y; h[q*4+2] = hq.z; h[q*4+3] = hq.w;
    }

    const float Dch = Dvec[ch];
    const float* zrow = Z + ((size_t)b * D_INN + ch) * SEQLEN + (size_t)c * LC;

    for (int t = 0; t < LC; ++t) {
        const size_t off  = (row0 + t) * D_INN + ch;
        if (t + 8 < LC) {
            __builtin_prefetch(&Delta[off + 8 * D_INN], 0, 0);
            __builtin_prefetch(&X[off + 8 * D_INN], 0, 0);
        }
        const float delta = Delta[off];
        const float xv    = X[off];
        const float u     = delta * xv;
        float y = 0.f;
        const float4* Bv = (const float4*)(sB + t * NSTATE);
        const float4* Cv = (const float4*)(sC + t * NSTATE);
#pragma unroll
        for (int q = 0; q < 4; ++q) {
            float4 bq = Bv[q], cq = Cv[q];
            float bb[4] = {bq.x, bq.y, bq.z, bq.w};
            float cc[4] = {cq.x, cq.y, cq.z, cq.w};
#pragma unroll
            for (int j = 0; j < 4; ++j) {
                const int n = q * 4 + j;
                const float dA = __expf(delta * Aa[n]);
                h[n] = dA * h[n] + u * bb[j];
                y   += h[n] * cc[j];
            }
        }
        y += xv * Dch;
        const float zl  = zrow[t];
        const float sig = 1.0f / (1.0f + __expf(-zl));
        __builtin_nontemporal_store(y * (zl * sig), &Out[off]);
    }
}

// ---------------------------------------------------------------------------
// Launch
// ---------------------------------------------------------------------------
extern "C" void kernel_launch(void* const* d_in, const int* in_sizes, int n_in,
                              void* d_out, int out_size, void* d_ws, size_t ws_size,
                              hipStream_t stream)
{
    const float* x     = (const float*)d_in[0];  // (2,2048,768)
    const float* z     = (const float*)d_in[1];  // (2,768,2048)
    const float* A_log = (const float*)d_in[2];  // (768,16)
    const float* Dvec  = (const float*)d_in[3];  // (768,)
    const float* Wx    = (const float*)d_in[4];  // (1680,768)
    const float* Wdt   = (const float*)d_in[5];  // (768,48)
    const float* bdt   = (const float*)d_in[6];  // (768,)
    float* out = (float*)d_out;

    char* ws = (char*)d_ws;
    const size_t OFF_P80   = 0;
    const size_t OFF_DELTA = OFF_P80   + (size_t)M_TOTAL * NPROJ * 4;
    const size_t OFF_PPROD = OFF_DELTA + (size_t)M_TOTAL * D_INN * 4;
    const size_t OFF_HEND  = OFF_PPROD + (size_t)BATCH * NCHUNK * D_INN * NSTATE * 4;
    const size_t OFF_HIN   = OFF_HEND  + (size_t)BATCH * NCHUNK * D_INN * NSTATE * 4;

    float* P80   = (float*)(ws + OFF_P80);
    float* Delta = (float*)(ws + OFF_DELTA);
    float* Pprod = (float*)(ws + OFF_PPROD);
    float* Hend  = (float*)(ws + OFF_HEND);
    float* Hin   = (float*)(ws + OFF_HIN);

    // 1) x @ Wsub^T  (only the 80 live columns of the 1680-wide projection)
    k_proj_gemm<<<dim3(M_TOTAL / 16), 32, 0, stream>>>(x, Wx, P80);

    // 2) delta = softplus(dt @ W_dt^T + 2*b_dt)
    k_delta_gemm<<<dim3(M_TOTAL / 16, D_INN / 16), 32, 0, stream>>>(P80, Wdt, bdt, Delta);

    // 3) chunked parallel scan
    k_scan_local  <<<dim3(NCHUNK, BATCH), D_INN, 0, stream>>>(x, Delta, P80, A_log, Pprod, Hend);
    k_scan_combine<<<dim3((BATCH * D_INN * NSTATE) / 256), 256, 0, stream>>>(Pprod, Hend, Hin);
    k_scan_final  <<<dim3(NCHUNK, BATCH), D_INN, 0, stream>>>(x, Delta, P80, A_log, Dvec, z, Hin, out);
}